// PoseCost_55319178772544
// MI455X (gfx1250) — compile-verified
//
#include <hip/hip_runtime.h>
#include <stdint.h>

#define BLK 256

// ---- CDNA5 async global->LDS copy (ASYNCcnt-tracked DMA into LDS) ----------
// LDS destination is byte offset lds_off; 16 bytes per enabled lane, fully coalesced.
__device__ __forceinline__ void async_g2lds_b128(uint32_t lds_off, const void* g) {
  asm volatile("global_load_async_to_lds_b128 %0, %1, off"
               :: "v"(lds_off), "v"((unsigned long long)(uintptr_t)g)
               : "memory");
}

__device__ __forceinline__ void wait_asynccnt0() {
  asm volatile("s_wait_asynccnt 0x0" ::: "memory");
}

// ---- Read one element's staged data from LDS: 12x ds_load_b32 + ONE wait ---
// raddr/paddr are LDS byte offsets (addrspace(3) values). Explicit asm keeps
// the LDS path live (compiler cannot see the async DMA writes) and avoids the
// per-access waits that volatile flat loads incur.
__device__ __forceinline__ void lds_read_elem(
    uint32_t raddr, uint32_t paddr,
    float& r0, float& r1, float& r2, float& r3, float& r4, float& r5,
    float& r6, float& r7, float& r8, float& p0, float& p1, float& p2) {
  asm volatile(
      "ds_load_b32 %0, %12 offset:0\n\t"
      "ds_load_b32 %1, %12 offset:4\n\t"
      "ds_load_b32 %2, %12 offset:8\n\t"
      "ds_load_b32 %3, %12 offset:12\n\t"
      "ds_load_b32 %4, %12 offset:16\n\t"
      "ds_load_b32 %5, %12 offset:20\n\t"
      "ds_load_b32 %6, %12 offset:24\n\t"
      "ds_load_b32 %7, %12 offset:28\n\t"
      "ds_load_b32 %8, %12 offset:32\n\t"
      "ds_load_b32 %9, %13 offset:0\n\t"
      "ds_load_b32 %10, %13 offset:4\n\t"
      "ds_load_b32 %11, %13 offset:8\n\t"
      "s_wait_dscnt 0x0"
      : "=&v"(r0), "=&v"(r1), "=&v"(r2), "=&v"(r3), "=&v"(r4), "=&v"(r5),
        "=&v"(r6), "=&v"(r7), "=&v"(r8), "=&v"(p0), "=&v"(p1), "=&v"(p2)
      : "v"(raddr), "v"(paddr)
      : "memory");
}

__global__ __launch_bounds__(BLK) void pose_cost_kernel(
    const float* __restrict__ ee_pos,   // [N,3]
    const float* __restrict__ ee_rot,   // [N,3,3]
    const float* __restrict__ gpos,     // [3]
    const float* __restrict__ grot,     // [3,3]
    const float* __restrict__ vw,       // [6]
    float* __restrict__ out,            // cost[N] | rot_err[N] | pos_err[N] | v[N,3] | omega[N,3]
    int n)
{
  __shared__ __align__(16) float sR[BLK * 9];  // 9216 B
  __shared__ __align__(16) float sP[BLK * 3];  // 3072 B

  const int tid = threadIdx.x;
  const long long base = (long long)blockIdx.x * BLK;
  const long long idx  = base + tid;
  const bool full = (base + BLK) <= (long long)n;

  // LDS byte offsets of the static shared arrays (low 32 bits of the flat
  // address == addrspace(3) offset on AMDGCN).
  const uint32_t r_off = (uint32_t)(uintptr_t)(void*)sR;
  const uint32_t p_off = (uint32_t)(uintptr_t)(void*)sP;

  if (full) {
    const float* rsrc = ee_rot + base * 9;   // 2304 floats = 576 float4 (16B aligned)
    const float* psrc = ee_pos + base * 3;   // 768 floats  = 192 float4 (16B aligned)

    async_g2lds_b128(r_off + (uint32_t)tid * 16u,         rsrc + (long long)tid * 4);
    async_g2lds_b128(r_off + (uint32_t)(tid + 256) * 16u, rsrc + (long long)(tid + 256) * 4);
    if (tid < 64)   // waves 0..1 only -> uniform per wave, no divergence
      async_g2lds_b128(r_off + (uint32_t)(tid + 512) * 16u, rsrc + (long long)(tid + 512) * 4);
    if (tid < 192)  // waves 0..5 only
      async_g2lds_b128(p_off + (uint32_t)tid * 16u, psrc + (long long)tid * 4);

    wait_asynccnt0();   // each wave waits for its own DMAs ...
  }
  __syncthreads();      // ... barrier makes all waves' LDS writes visible

  if (idx >= (long long)n) return;

  // ---- uniform goal data (scalar loads) ----
  const float G0 = grot[0], G1 = grot[1], G2 = grot[2];
  const float G3 = grot[3], G4 = grot[4], G5 = grot[5];
  const float G6 = grot[6], G7 = grot[7], G8 = grot[8];
  const float gp0 = gpos[0], gp1 = gpos[1], gp2 = gpos[2];
  const float w0 = vw[0], w1 = vw[1], w2w = vw[2];
  const float w3 = vw[3], w4 = vw[4], w5 = vw[5];

  // ---- per-element data ----
  float r0, r1, r2, r3, r4, r5, r6, r7, r8, p0, p1, p2;
  if (full) {
    lds_read_elem(r_off + (uint32_t)tid * 36u, p_off + (uint32_t)tid * 12u,
                  r0, r1, r2, r3, r4, r5, r6, r7, r8, p0, p1, p2);
  } else {
    const float* rp = ee_rot + idx * 9;
    r0 = rp[0]; r1 = rp[1]; r2 = rp[2];
    r3 = rp[3]; r4 = rp[4]; r5 = rp[5];
    r6 = rp[6]; r7 = rp[7]; r8 = rp[8];
    const float* pp = ee_pos + idx * 3;
    p0 = pp[0]; p1 = pp[1]; p2 = pp[2];
  }

  // ---- t_ge = R^T (gp - p) ;  Rg = R^T G  (r[3k+i] is R_{k,i}) ----
  const float d0 = gp0 - p0, d1 = gp1 - p1, d2 = gp2 - p2;
  const float t0 = r0 * d0 + r3 * d1 + r6 * d2;
  const float t1 = r1 * d0 + r4 * d1 + r7 * d2;
  const float t2 = r2 * d0 + r5 * d1 + r8 * d2;

  const float Rg00 = r0 * G0 + r3 * G3 + r6 * G6;
  const float Rg01 = r0 * G1 + r3 * G4 + r6 * G7;
  const float Rg02 = r0 * G2 + r3 * G5 + r6 * G8;
  const float Rg10 = r1 * G0 + r4 * G3 + r7 * G6;
  const float Rg11 = r1 * G1 + r4 * G4 + r7 * G7;
  const float Rg12 = r1 * G2 + r4 * G5 + r7 * G8;
  const float Rg20 = r2 * G0 + r5 * G3 + r8 * G6;
  const float Rg21 = r2 * G1 + r5 * G4 + r8 * G7;
  const float Rg22 = r2 * G2 + r5 * G5 + r8 * G8;

  // ---- SE(3) log map ----
  const float tr = Rg00 + Rg11 + Rg22;
  float ct = (tr - 1.0f) * 0.5f;
  ct = fminf(fmaxf(ct, -1.0f + 1e-7f), 1.0f - 1e-7f);
  const float th = acosf(ct);
  const float st = sinf(th);

  const float wv0 = 0.5f * (Rg21 - Rg12);
  const float wv1 = 0.5f * (Rg02 - Rg20);
  const float wv2 = 0.5f * (Rg10 - Rg01);

  const bool small = th < 1e-4f;
  const float safe_sin = small ? 1.0f : st;
  const float scale = small ? (1.0f + th * th * (1.0f / 6.0f)) : (th / safe_sin);
  const float o0 = scale * wv0, o1 = scale * wv1, o2 = scale * wv2;

  const float th2 = small ? 1.0f : th * th;
  const float A = small ? (1.0f / 12.0f)
                        : (1.0f - (th * st) / (2.0f * (1.0f - ct))) / th2;

  // V^{-1} t = (1 - A|w|^2) t - 0.5 (w x t) + A (w . t) w   [uses (wx)^2 = ww^T - |w|^2 I]
  const float osq = o0 * o0 + o1 * o1 + o2 * o2;
  const float odt = o0 * t0 + o1 * t1 + o2 * t2;
  const float c0 = o1 * t2 - o2 * t1;
  const float c1 = o2 * t0 - o0 * t2;
  const float c2 = o0 * t1 - o1 * t0;
  const float s1 = 1.0f - A * osq;
  const float Ad = A * odt;
  const float v0 = s1 * t0 - 0.5f * c0 + Ad * o0;
  const float v1 = s1 * t1 - 0.5f * c1 + Ad * o1;
  const float v2 = s1 * t2 - 0.5f * c2 + Ad * o2;

  // ---- costs (CONV thresholds are 0.0 and errs are >= 0 -> where() is identity) ----
  const float ro0 = w0 * o0, ro1 = w1 * o1, ro2 = w2w * o2;
  const float rotation_err = ro0 * ro0 + ro1 * ro1 + ro2 * ro2;
  const float pv0 = w3 * v0, pv1 = w4 * v1, pv2 = w5 * v2;
  const float position_err = pv0 * pv0 + pv1 * pv1 + pv2 * pv2;
  const float cost = 15.0f * rotation_err + 100.0f * position_err;

  // ---- outputs (streaming, never re-read -> non-temporal stores) ----
  const long long N = (long long)n;
  __builtin_nontemporal_store(cost,         out + idx);
  __builtin_nontemporal_store(rotation_err, out + N + idx);
  __builtin_nontemporal_store(position_err, out + 2 * N + idx);
  float* vout = out + 3 * N + idx * 3;
  __builtin_nontemporal_store(v0, vout + 0);
  __builtin_nontemporal_store(v1, vout + 1);
  __builtin_nontemporal_store(v2, vout + 2);
  float* oout = out + 6 * N + idx * 3;
  __builtin_nontemporal_store(o0, oout + 0);
  __builtin_nontemporal_store(o1, oout + 1);
  __builtin_nontemporal_store(o2, oout + 2);
}

extern "C" void kernel_launch(void* const* d_in, const int* in_sizes, int n_in,
                              void* d_out, int out_size, void* d_ws, size_t ws_size,
                              hipStream_t stream) {
  const float* ee_pos = (const float*)d_in[0];  // [N,3]
  const float* ee_rot = (const float*)d_in[1];  // [N,3,3]
  const float* gpos   = (const float*)d_in[2];  // [3]
  const float* grot   = (const float*)d_in[3];  // [3,3]
  const float* vw     = (const float*)d_in[4];  // [6]
  float* out = (float*)d_out;

  const long long n = (long long)in_sizes[0] / 3;   // B*H elements
  const int blocks = (int)((n + BLK - 1) / BLK);

  pose_cost_kernel<<<blocks, BLK, 0, stream>>>(ee_pos, ee_rot, gpos, grot, vw,
                                               out, (int)n);
}